// ClusterGCNEdges_20968030339124
// MI455X (gfx1250) — compile-verified
//
#include <hip/hip_runtime.h>
#include <math.h>

typedef __attribute__((ext_vector_type(2))) float v2f;
typedef __attribute__((ext_vector_type(8))) float v8f;

#define NTHREADS 256

// ---------------------------------------------------------------------------
// zero-fill (grid-stride)
// ---------------------------------------------------------------------------
__global__ void fill_zero_f32(float* __restrict__ p, long long n) {
    long long i = (long long)blockIdx.x * blockDim.x + threadIdx.x;
    long long stride = (long long)gridDim.x * blockDim.x;
    for (; i < n; i += stride) p[i] = 0.0f;
}

// ---------------------------------------------------------------------------
// per-destination edge counts (float -> native f32 atomics)
// ---------------------------------------------------------------------------
__global__ void count_edges(const int* __restrict__ dst, float* __restrict__ cnt, int E) {
    int i = blockIdx.x * blockDim.x + threadIdx.x;
    if (i < E) unsafeAtomicAdd(&cnt[dst[i]], 1.0f);
}

__global__ void invert_counts(float* __restrict__ cnt, int n) {
    int i = blockIdx.x * blockDim.x + threadIdx.x;
    if (i < n) cnt[i] = 1.0f / fmaxf(cnt[i], 1.0f);
}

// ---------------------------------------------------------------------------
// scatter: agg[dst] += w[e] * x[src], float4 per thread, chunk-per-thread
// mapping so consecutive lanes hit consecutive features of the same row
// (coalesced gather + coalesced atomics). shift = log2(d/4).
// ---------------------------------------------------------------------------
__global__ void scatter_weighted(const float* __restrict__ x,
                                 const int* __restrict__ src,
                                 const int* __restrict__ dst,
                                 const float* __restrict__ w,
                                 float* __restrict__ agg,
                                 int E, int d, int shift) {
    long long tid = (long long)blockIdx.x * blockDim.x + threadIdx.x;
    long long total = (long long)E << shift;
    if (tid >= total) return;
    int e = (int)(tid >> shift);
    int c = (int)(tid & ((1 << shift) - 1));
    int s = src[e];
    int t = dst[e];
    float ww = w[e];
    const float4* xv = (const float4*)(x + (size_t)s * d);
    float4 v = xv[c];
    float* base = agg + (size_t)t * d + c * 4;
    unsafeAtomicAdd(base + 0, v.x * ww);
    unsafeAtomicAdd(base + 1, v.y * ww);
    unsafeAtomicAdd(base + 2, v.z * ww);
    unsafeAtomicAdd(base + 3, v.w * ww);
}

// ---------------------------------------------------------------------------
// Fused SAGE GEMM via V_WMMA_F32_16X16X4_F32 (wave32, one wave per 16x16 tile)
//   out = (agg * inv) @ Wl + x @ Wr + (bl + br)   [+ ReLU]
//
// A 16x4 layout: lanes0-15 M=0..15; vgpr0 K=half*2, vgpr1 K=half*2+1.
// B 4x16 layout mirrors A; C/D standard 8-VGPR f32 layout.
//
// Column masking trick: C column n depends only on B column n, so lanes with
// bcol >= d_out just compute dead columns. We CLAMP the load address (stays
// in-bounds) and load unconditionally -> fully branchless K-loop, no EXEC
// save/restore. The store guard in the epilogue drops the dead columns.
// ---------------------------------------------------------------------------
__global__ void sage_gemm_wmma(const float* __restrict__ agg,
                               const float* __restrict__ inv,
                               const float* __restrict__ xin,
                               const float* __restrict__ Wl,
                               const float* __restrict__ bl,
                               const float* __restrict__ Wr,
                               const float* __restrict__ br,
                               float* __restrict__ out,
                               int n_nodes, int d_in, int d_out,
                               int tiles_n, int do_relu) {
    int lane = threadIdx.x & 31;
    int wid = blockIdx.x * (blockDim.x >> 5) + (threadIdx.x >> 5);
    int tiles_m = n_nodes >> 4;   // 50000 / 16 = 3125 exact
    if (wid >= tiles_m * tiles_n) return;
    int tm = wid / tiles_n;
    int tn = wid - tm * tiles_n;
    int m0 = tm << 4;
    int n0 = tn << 4;

    int half = lane >> 4;        // 0: K={0,1}, C rows 0-7 ; 1: K={2,3}, C rows 8-15
    int l15  = lane & 15;
    int arow = m0 + l15;         // A row this lane feeds
    int bcol = n0 + l15;         // B/output column this lane feeds
    bool colok = bcol < d_out;
    int bc = colok ? bcol : (d_out - 1);  // clamped, always in-bounds

    float scale = inv[arow];
    const float* aggRow = agg + (size_t)arow * d_in + half * 2;
    const float* xRow   = xin + (size_t)arow * d_in + half * 2;
    const float* WlCol  = Wl + (size_t)(half * 2) * d_out + bc;
    const float* WrCol  = Wr + (size_t)(half * 2) * d_out + bc;

    v8f c = {};

    // pass 1: mean @ Wl  (mean = agg * inv, fused into the A load)
    for (int k = 0; k < d_in; k += 4) {
        v2f a = *(const v2f*)(aggRow + k);
        a.x *= scale;
        a.y *= scale;
        v2f b;
        b.x = WlCol[(size_t)k * d_out];
        b.y = WlCol[(size_t)(k + 1) * d_out];
        c = __builtin_amdgcn_wmma_f32_16x16x4_f32(false, a, false, b,
                                                  (short)0, c, false, false);
    }
    // pass 2: x @ Wr, accumulated into the same tile
    for (int k = 0; k < d_in; k += 4) {
        v2f a = *(const v2f*)(xRow + k);
        v2f b;
        b.x = WrCol[(size_t)k * d_out];
        b.y = WrCol[(size_t)(k + 1) * d_out];
        c = __builtin_amdgcn_wmma_f32_16x16x4_f32(false, a, false, b,
                                                  (short)0, c, false, false);
    }

    float bias = bl[bc] + br[bc];   // dead lanes read a valid column; value unused
    if (colok) {
        for (int r = 0; r < 8; ++r) {
            float v = c[r] + bias;
            if (do_relu) v = fmaxf(v, 0.0f);
            int row = m0 + r + half * 8;
            out[(size_t)row * d_out + bcol] = v;
        }
    }
}

// ---------------------------------------------------------------------------
// in-place log_softmax over 8 columns, one thread per row
// ---------------------------------------------------------------------------
__global__ void log_softmax8(float* __restrict__ out, int n) {
    int i = blockIdx.x * blockDim.x + threadIdx.x;
    if (i >= n) return;
    float v[8];
    float m = -INFINITY;
    float* row = out + (size_t)i * 8;
#pragma unroll
    for (int j = 0; j < 8; ++j) { v[j] = row[j]; m = fmaxf(m, v[j]); }
    float s = 0.0f;
#pragma unroll
    for (int j = 0; j < 8; ++j) s += __expf(v[j] - m);
    float ls = __logf(s);
#pragma unroll
    for (int j = 0; j < 8; ++j) row[j] = v[j] - m - ls;
}

// ---------------------------------------------------------------------------
// host-side orchestration
// ---------------------------------------------------------------------------
static inline int cdiv(long long a, long long b) { return (int)((a + b - 1) / b); }

extern "C" void kernel_launch(void* const* d_in, const int* in_sizes, int n_in,
                              void* d_out, int out_size, void* d_ws, size_t ws_size,
                              hipStream_t stream) {
    const float* x    = (const float*)d_in[0];
    const int*   ei   = (const int*)d_in[1];
    const float* w    = (const float*)d_in[2];
    const float* Wl0  = (const float*)d_in[3];
    const float* bl0  = (const float*)d_in[4];
    const float* Wr0  = (const float*)d_in[5];
    const float* br0  = (const float*)d_in[6];
    const float* Wl1  = (const float*)d_in[7];
    const float* bl1  = (const float*)d_in[8];
    const float* Wr1  = (const float*)d_in[9];
    const float* br1  = (const float*)d_in[10];
    const float* Wl2  = (const float*)d_in[11];
    const float* bl2  = (const float*)d_in[12];
    const float* Wr2  = (const float*)d_in[13];
    const float* br2  = (const float*)d_in[14];

    const int D0 = 128, D1 = 256, D2 = 256;  // layer input dims
    const int N = in_sizes[0] / D0;          // 50000
    const int E = in_sizes[2];               // 800000
    const int* src = ei;
    const int* dst = ei + E;

    // workspace layout (floats): inv[N] | agg[N*256] | hA[N*256] | hB[N*256]
    float* inv = (float*)d_ws;
    float* agg = inv + N;
    float* hA  = agg + (size_t)N * 256;
    float* hB  = hA  + (size_t)N * 256;
    float* logits = (float*)d_out;           // [N, 8]

    const int T = NTHREADS;
    const int fillBlocks = 2048;

    // --- edge counts -> inv = 1/max(cnt,1), computed once (graph-invariant)
    fill_zero_f32<<<fillBlocks, T, 0, stream>>>(inv, N);
    count_edges<<<cdiv(E, T), T, 0, stream>>>(dst, inv, E);
    invert_counts<<<cdiv(N, T), T, 0, stream>>>(inv, N);

    // --- layer 0: x[N,128] -> hA[N,256], ReLU
    fill_zero_f32<<<fillBlocks, T, 0, stream>>>(agg, (long long)N * D0);
    scatter_weighted<<<cdiv((long long)E * (D0 / 4), T), T, 0, stream>>>(
        x, src, dst, w, agg, E, D0, 5);
    {
        int tiles_n = 256 / 16;
        int waves = (N / 16) * tiles_n;
        sage_gemm_wmma<<<cdiv(waves, T / 32), T, 0, stream>>>(
            agg, inv, x, Wl0, bl0, Wr0, br0, hA, N, D0, 256, tiles_n, 1);
    }

    // --- layer 1: hA[N,256] -> hB[N,256], ReLU
    fill_zero_f32<<<fillBlocks, T, 0, stream>>>(agg, (long long)N * D1);
    scatter_weighted<<<cdiv((long long)E * (D1 / 4), T), T, 0, stream>>>(
        hA, src, dst, w, agg, E, D1, 6);
    {
        int tiles_n = 256 / 16;
        int waves = (N / 16) * tiles_n;
        sage_gemm_wmma<<<cdiv(waves, T / 32), T, 0, stream>>>(
            agg, inv, hA, Wl1, bl1, Wr1, br1, hB, N, D1, 256, tiles_n, 1);
    }

    // --- layer 2: hB[N,256] -> logits[N,8] (one 16-col tile, cols 8..15 dead)
    fill_zero_f32<<<fillBlocks, T, 0, stream>>>(agg, (long long)N * D2);
    scatter_weighted<<<cdiv((long long)E * (D2 / 4), T), T, 0, stream>>>(
        hB, src, dst, w, agg, E, D2, 6);
    {
        int tiles_n = 1;
        int waves = (N / 16) * tiles_n;
        sage_gemm_wmma<<<cdiv(waves, T / 32), T, 0, stream>>>(
            agg, inv, hB, Wl2, bl2, Wr2, br2, logits, N, D2, 8, tiles_n, 0);
    }

    // --- final log_softmax over 8 classes, in place on d_out
    log_softmax8<<<cdiv(N, T), T, 0, stream>>>(logits, N);
}